// DiffusionNetAutoencoder_45904610460386
// MI455X (gfx1250) — compile-verified
//
#include <hip/hip_runtime.h>
#include <math.h>

// ---------------------------------------------------------------------------
// CDNA5 (gfx1250) SpiralNet decoder.
// All matmuls run on V_WMMA_F32_16X16X4_F32 (fp32 in, fp32 accumulate) to
// match the fp32 reference numerics while engaging the matrix pipe.
// One wave32 owns one 16-row M-tile and accumulates ALL N-tiles (NT<=8)
// so each gathered A fragment (and its spiral address math) feeds NT WMMAs.
// ---------------------------------------------------------------------------

typedef float v2f __attribute__((ext_vector_type(2)));
typedef float v8f __attribute__((ext_vector_type(8)));

#define WPB 4  // waves per block

// ---------------- zero fill ----------------
__global__ void zero_f32(float* __restrict__ p, long long n) {
  long long i = (long long)blockIdx.x * blockDim.x + threadIdx.x;
  long long stride = (long long)gridDim.x * blockDim.x;
  for (; i < n; i += stride) p[i] = 0.0f;
}

// ---------------- sparse up-transform: out[:,rows[j],:] += x[:,cols[j],:]*vals[j]
__global__ void pool_scatter(const float* __restrict__ x,   // [T, nc, C]
                             const int*   __restrict__ rows,
                             const int*   __restrict__ cols,
                             const float* __restrict__ vals,
                             float* __restrict__ out,        // [T, nf, C] (pre-zeroed)
                             int nnz, int nc, int nf, int C) {
  long long total = 64LL * nnz * C;
  long long i = (long long)blockIdx.x * blockDim.x + threadIdx.x;
  long long stride = (long long)gridDim.x * blockDim.x;
  for (; i < total; i += stride) {
    int c = (int)(i % C);
    long long jj = i / C;
    int j = (int)(jj % nnz);
    int t = (int)(jj / nnz);
    float g = x[((size_t)t * nc + cols[j]) * C + c] * vals[j];
    atomicAdd(out + ((size_t)t * nf + rows[j]) * C + c, g);  // global_atomic_add_f32
  }
}

// ---------------- dense0: out[M,N] = A[M,K] @ W[N,K]^T + bias (M,N mult of 16)
__global__ void dense_wmma(const float* __restrict__ A,
                           const float* __restrict__ W,
                           const float* __restrict__ bias,
                           float* __restrict__ out,
                           int M, int N, int K) {
  int wave = threadIdx.x >> 5;
  int lane = threadIdx.x & 31;
  int nTiles = N >> 4;
  int tile = blockIdx.x * WPB + wave;
  if (tile >= (M >> 4) * nTiles) return;          // wave-uniform
  int nT = tile % nTiles, mT = tile / nTiles;
  int aRow = (mT << 4) + (lane & 15);             // A-frag row (both half-waves)
  int col  = (nT << 4) + (lane & 15);             // B-frag / C column
  int koff = (lane < 16) ? 0 : 2;                 // K sub-offset per half-wave
  const float* Ar = A + (size_t)aRow * K;
  const float* Wc = W + (size_t)col * K;
  v8f acc = {};
  for (int k = 0; k < K; k += 4) {
    v2f a = *(const v2f*)(Ar + k + koff);
    v2f b = *(const v2f*)(Wc + k + koff);
    acc = __builtin_amdgcn_wmma_f32_16x16x4_f32(false, a, false, b,
                                                (short)0, acc, false, false);
  }
  float bc = bias[col];
  int rbase = (mT << 4) + ((lane >> 4) << 3);     // C/D: lanes>=16 hold M=r+8
#pragma unroll
  for (int r = 0; r < 8; ++r)
    out[(size_t)(rbase + r) * N + col] = acc[r] + bc;
}

// ---------------- fused spiral gather + GEMM + bias (+ELU) (+actor) ----------
// x: [T=64, n, Cin]; spiral: [n, 9]; W: [Cout, 9*Cin]; out: [T, n, Cout]
// NT = number of 16-wide N-tiles this wave accumulates (== ceil(Cout/16)).
// MASK = true only when Cout is not a multiple of 16 (final Cout=3 conv).
template <int NT, bool MASK>
__global__ void spiral_conv_wmma(const float* __restrict__ x,
                                 const int*   __restrict__ spiral,
                                 const float* __restrict__ W,
                                 const float* __restrict__ bias,
                                 const float* __restrict__ actor,  // [n,3] or null
                                 float* __restrict__ out,
                                 int n, int Cin, int Cout, int doElu) {
  const int S = 9;
  const int K = S * Cin;
  int wave = threadIdx.x >> 5;
  int lane = threadIdx.x & 31;
  int Mtotal = 64 * n;
  int Mtiles = (Mtotal + 15) >> 4;
  int mT = blockIdx.x * WPB + wave;
  if (mT >= Mtiles) return;                        // wave-uniform

  // A-fragment row for this lane (clamped; padded rows recomputed but unstored)
  int aRow = (mT << 4) + (lane & 15);
  if (aRow >= Mtotal) aRow = Mtotal - 1;
  int t = aRow / n, v = aRow - t * n;
  const int* spv = spiral + (size_t)v * S;
  const float* xt = x + (size_t)t * n * Cin;
  int koff = (lane < 16) ? 0 : 2;                  // K sub-offset per half-wave
  int lc = lane & 15;

  // per-N-tile weight row pointers (+ branch-free column mask when needed)
  const float* Wc[NT];
  float cmask[NT];
#pragma unroll
  for (int j = 0; j < NT; ++j) {
    int col = (j << 4) + lc;
    if (MASK) {
      bool ok = col < Cout;
      cmask[j] = ok ? 1.0f : 0.0f;
      Wc[j] = W + (size_t)(ok ? col : (Cout - 1)) * K;
    } else {
      cmask[j] = 1.0f;
      Wc[j] = W + (size_t)col * K;
    }
  }

  v8f acc[NT];
#pragma unroll
  for (int j = 0; j < NT; ++j)
#pragma unroll
    for (int r = 0; r < 8; ++r) acc[j][r] = 0.0f;

  for (int s = 0; s < S; ++s) {
    const float* xrow = xt + (size_t)spv[s] * Cin;  // spiral gather
    int kbase = s * Cin;
    for (int c = 0; c < Cin; c += 4) {
      v2f a = *(const v2f*)(xrow + c + koff);       // one A frag feeds NT WMMAs
#pragma unroll
      for (int j = 0; j < NT; ++j) {
        v2f b = *(const v2f*)(Wc[j] + kbase + c + koff);
        if (MASK) { b.x *= cmask[j]; b.y *= cmask[j]; }  // no EXEC change pre-WMMA
        acc[j] = __builtin_amdgcn_wmma_f32_16x16x4_f32(false, a, false, b,
                                                       (short)0, acc[j],
                                                       false, false);
      }
    }
  }

  int rbase = (mT << 4) + ((lane >> 4) << 3);       // C/D: lanes>=16 hold M=r+8
#pragma unroll
  for (int j = 0; j < NT; ++j) {
    int col = (j << 4) + lc;
    if (!MASK || col < Cout) {
      float bc = bias[col];
#pragma unroll
      for (int r = 0; r < 8; ++r) {
        int row = rbase + r;
        if (row < Mtotal) {
          float val = acc[j][r] + bc;
          if (doElu) val = (val > 0.0f) ? val : (expf(val) - 1.0f);
          if (actor) val += actor[(size_t)(row % n) * 3 + col];
          out[(size_t)row * Cout + col] = val;      // row == t*n + v
        }
      }
    }
  }
}

// ---------------------------------------------------------------------------
extern "C" void kernel_launch(void* const* d_in, const int* in_sizes, int n_in,
                              void* d_out, int out_size, void* d_ws, size_t ws_size,
                              hipStream_t stream) {
  (void)in_sizes; (void)n_in; (void)out_size; (void)ws_size;

  const float* latent  = (const float*)d_in[0];
  const float* actor   = (const float*)d_in[1];
  const int* spiral0   = (const int*)d_in[2];
  const int* spiral1   = (const int*)d_in[3];
  const int* spiral2   = (const int*)d_in[4];
  const int* spiral3   = (const int*)d_in[5];
  const int* up0_rows  = (const int*)d_in[6];
  const int* up0_cols  = (const int*)d_in[7];
  const float* up0_vals = (const float*)d_in[8];
  const int* up1_rows  = (const int*)d_in[9];
  const int* up1_cols  = (const int*)d_in[10];
  const float* up1_vals = (const float*)d_in[11];
  const int* up2_rows  = (const int*)d_in[12];
  const int* up2_cols  = (const int*)d_in[13];
  const float* up2_vals = (const float*)d_in[14];
  const int* up3_rows  = (const int*)d_in[15];
  const int* up3_cols  = (const int*)d_in[16];
  const float* up3_vals = (const float*)d_in[17];
  const float* W0 = (const float*)d_in[18]; const float* b0 = (const float*)d_in[19];
  const float* W1 = (const float*)d_in[20]; const float* b1 = (const float*)d_in[21];
  const float* W2 = (const float*)d_in[22]; const float* b2 = (const float*)d_in[23];
  const float* W3 = (const float*)d_in[24]; const float* b3 = (const float*)d_in[25];
  const float* W4 = (const float*)d_in[26]; const float* b4 = (const float*)d_in[27];
  const float* Wf = (const float*)d_in[28]; const float* bf = (const float*)d_in[29];

  // ping-pong workspace: pooled buffers (bufB, max 64*5023*64 f32 = 82.3MB),
  // conv/dense outputs (bufA, max 64*5023*32 f32 = 41.1MB)
  float* bufB = (float*)d_ws;
  float* bufA = (float*)((char*)d_ws + 82296832);  // 64*5023*64*4, 256B aligned

  auto zero = [&](float* p, long long cnt) {
    zero_f32<<<2048, 256, 0, stream>>>(p, cnt);
  };
  auto pool = [&](const float* x, const int* r, const int* c, const float* vls,
                  float* o, int nnz, int nc, int nf, int C) {
    pool_scatter<<<4096, 256, 0, stream>>>(x, r, c, vls, o, nnz, nc, nf, C);
  };
  auto conv = [&](const float* x, const int* sp, const float* W, const float* b,
                  const float* act, float* o, int n, int Cin, int Cout, int elu) {
    int Mtiles = (64 * n + 15) / 16;
    int blocks = (Mtiles + WPB - 1) / WPB;
    int Ntiles = (Cout + 15) / 16;
    switch (Ntiles) {
      case 8:
        spiral_conv_wmma<8, false><<<blocks, 32 * WPB, 0, stream>>>(
            x, sp, W, b, act, o, n, Cin, Cout, elu);
        break;
      case 4:
        spiral_conv_wmma<4, false><<<blocks, 32 * WPB, 0, stream>>>(
            x, sp, W, b, act, o, n, Cin, Cout, elu);
        break;
      case 2:
        spiral_conv_wmma<2, false><<<blocks, 32 * WPB, 0, stream>>>(
            x, sp, W, b, act, o, n, Cin, Cout, elu);
        break;
      default:
        spiral_conv_wmma<1, true><<<blocks, 32 * WPB, 0, stream>>>(
            x, sp, W, b, act, o, n, Cin, Cout, elu);
        break;
    }
  };

  // 1) dense0: [64,128] @ [2560,128]^T -> bufA = x0 [64,20,128]
  {
    int tiles = (64 / 16) * (2560 / 16);
    dense_wmma<<<(tiles + WPB - 1) / WPB, 32 * WPB, 0, stream>>>(
        latent, W0, b0, bufA, 64, 2560, 128);
  }
  // 2) level 3: pool 20->79 (C=128), conv 128->128
  zero(bufB, 64LL * 79 * 128);
  pool(bufA, up3_rows, up3_cols, up3_vals, bufB, 237, 20, 79, 128);
  conv(bufB, spiral3, W1, b1, nullptr, bufA, 79, 128, 128, 1);
  // 3) level 2: pool 79->314 (C=128), conv 128->64
  zero(bufB, 64LL * 314 * 128);
  pool(bufA, up2_rows, up2_cols, up2_vals, bufB, 942, 79, 314, 128);
  conv(bufB, spiral2, W2, b2, nullptr, bufA, 314, 128, 64, 1);
  // 4) level 1: pool 314->1256 (C=64), conv 64->64
  zero(bufB, 64LL * 1256 * 64);
  pool(bufA, up1_rows, up1_cols, up1_vals, bufB, 3768, 314, 1256, 64);
  conv(bufB, spiral1, W3, b3, nullptr, bufA, 1256, 64, 64, 1);
  // 5) level 0: pool 1256->5023 (C=64), conv 64->32
  zero(bufB, 64LL * 5023 * 64);
  pool(bufA, up0_rows, up0_cols, up0_vals, bufB, 15069, 1256, 5023, 64);
  conv(bufB, spiral0, W4, b4, nullptr, bufA, 5023, 64, 32, 1);
  // 6) final conv 32->3 (no ELU) + actor offset -> d_out [64,5023,3]
  conv(bufA, spiral0, Wf, bf, actor, (float*)d_out, 5023, 32, 3, 0);
}